// LightGCNConv_14216341750111
// MI455X (gfx1250) — compile-verified
//
#include <hip/hip_runtime.h>
#include <hip/hip_bf16.h>
#include <stdint.h>

#define D_FEAT 64

// gfx1250 async global->LDS path (ASYNCcnt-tracked). Guarded: falls back to
// direct loads + global_prefetch_b8 if the builtins are absent.
#if defined(__HIP_DEVICE_COMPILE__) && \
    __has_builtin(__builtin_amdgcn_global_load_async_to_lds_b64) && \
    __has_builtin(__builtin_amdgcn_s_wait_asynccnt)
#define USE_ASYNC_LDS 1
#else
#define USE_ASYNC_LDS 0
#endif

#if USE_ASYNC_LDS
typedef int v2i __attribute__((ext_vector_type(2)));
typedef __attribute__((address_space(1))) v2i* g64ptr;
typedef __attribute__((address_space(3))) v2i* l64ptr;
__device__ __forceinline__ void async_copy8(const float* g, float* l) {
  __builtin_amdgcn_global_load_async_to_lds_b64((g64ptr)(void*)g, (l64ptr)(void*)l, 0, 0);
}
#endif

__device__ __forceinline__ void atomic_add_f32_agent(float* p, float v) {
  // relaxed, agent scope -> hardware global_atomic_add_f32 (no return needed)
  __hip_atomic_fetch_add(p, v, __ATOMIC_RELAXED, __HIP_MEMORY_SCOPE_AGENT);
}

// Edge gather + scatter-add (defined FIRST so the disasm snippet shows it).
// One wave32 per edge; each lane owns a float2 of the 64-float row.
// All per-edge state (indices, norm) is wave-uniform -> forced into SGPRs via
// readfirstlane so index/norm loads become scalar (SMEM) loads and the output
// row base is SGPR math (saddr+voffset atomics).
__global__ __launch_bounds__(256) void scatter_kernel(
    const float* __restrict__ feat, const int* __restrict__ src, const int* __restrict__ dst,
    const float* __restrict__ nsrc, float* __restrict__ out, int E) {
  const int lane = threadIdx.x & 31;
  const int wib  = __builtin_amdgcn_readfirstlane(threadIdx.x >> 5);  // uniform by construction
  const int wave = blockIdx.x * (blockDim.x >> 5) + wib;
  const int nw   = gridDim.x * (blockDim.x >> 5);

#if USE_ASYNC_LDS
  __shared__ float lds[8 * 2 * D_FEAT];            // 2 row slots per wave = 4 KB/block
  float* pa = &lds[(wib * 2 + 0) * D_FEAT];        // current row slot
  float* pb = &lds[(wib * 2 + 1) * D_FEAT];        // next row slot

  int e = wave;
  if (e >= E) return;                              // wave-uniform
  int sCur = src[e];                               // scalar load (e uniform)
  async_copy8(&feat[(size_t)sCur * D_FEAT + lane * 2], &pa[lane * 2]);

  for (;;) {
    const int  en   = e + nw;
    const bool more = (en < E);
    int sNext = 0;
    if (more) {
      sNext = src[en];
      async_copy8(&feat[(size_t)sNext * D_FEAT + lane * 2], &pb[lane * 2]);
      __builtin_amdgcn_s_wait_asynccnt(1);         // current slot's copy done (in-order)
    } else {
      __builtin_amdgcn_s_wait_asynccnt(0);
    }
    const int   d   = dst[e];
    const float nsv = nsrc[sCur];
    const float2 f  = *(const float2*)&pa[lane * 2];
    float* orow = out + (size_t)d * D_FEAT + lane * 2;
    atomic_add_f32_agent(orow,     f.x * nsv);
    atomic_add_f32_agent(orow + 1, f.y * nsv);
    if (!more) break;
    e = en; sCur = sNext;
    float* t = pa; pa = pb; pb = t;                // swap slots
  }
#else
  for (int e = wave; e < E; e += nw) {
    const int epf = e + nw;
    if (epf < E)                                   // global_prefetch_b8 of next row
      __builtin_prefetch(&feat[(size_t)src[epf] * D_FEAT + lane * 2], 0, 0);
    const int s = src[e];
    const int d = dst[e];
    const float nsv = nsrc[s];
    const float2 f  = *(const float2*)&feat[(size_t)s * D_FEAT + lane * 2];
    float* orow = out + (size_t)d * D_FEAT + lane * 2;
    atomic_add_f32_agent(orow,     f.x * nsv);
    atomic_add_f32_agent(orow + 1, f.y * nsv);
  }
#endif
}

// Degree histograms (u32 hardware atomics; stay in L2).
__global__ void deg_kernel(const int* __restrict__ src, const int* __restrict__ dst,
                           unsigned* __restrict__ odeg, unsigned* __restrict__ ideg, int E) {
  int t = blockIdx.x * blockDim.x + threadIdx.x;
  if (t < E) {
    atomicAdd(&odeg[src[t]], 1u);
    atomicAdd(&ideg[dst[t]], 1u);
  }
}

// norms = clip(deg,1)^-0.5  (full-precision sqrt; only 100K threads).
__global__ void norm_kernel(const unsigned* __restrict__ odeg, const unsigned* __restrict__ ideg,
                            float* __restrict__ nsrc, float* __restrict__ ndst, int N) {
  int t = blockIdx.x * blockDim.x + threadIdx.x;
  if (t < N) {
    nsrc[t] = 1.0f / sqrtf(fmaxf((float)odeg[t], 1.0f));
    ndst[t] = 1.0f / sqrtf(fmaxf((float)ideg[t], 1.0f));
  }
}

// out[n,:] *= norm_dst[n]  (streaming, float2 per lane; norm load is uniform
// per wave-row so scalarize it too).
__global__ void scale_kernel(float* __restrict__ out, const float* __restrict__ ndst, int N) {
  long t = (long)blockIdx.x * blockDim.x + threadIdx.x;
  int n = (int)(t >> 5);
  int p = ((int)t & 31) * 2;
  if (n < N) {
    float v = ndst[__builtin_amdgcn_readfirstlane(n)];
    float2* o = (float2*)&out[(size_t)n * D_FEAT + p];
    float2 x = *o;
    x.x *= v; x.y *= v;
    *o = x;
  }
}

extern "C" void kernel_launch(void* const* d_in, const int* in_sizes, int n_in,
                              void* d_out, int out_size, void* d_ws, size_t ws_size,
                              hipStream_t stream) {
  (void)n_in; (void)ws_size;
  const float* feat = (const float*)d_in[0];
  const int*   src  = (const int*)d_in[1];
  const int*   dst  = (const int*)d_in[2];
  float*       out  = (float*)d_out;

  const int N = in_sizes[0] / D_FEAT;
  const int E = in_sizes[1];

  // Workspace layout: [odeg u32 N][ideg u32 N][nsrc f32 N][ndst f32 N] = 16*N bytes
  unsigned* odeg = (unsigned*)d_ws;
  unsigned* ideg = odeg + N;
  float*    nsrc = (float*)(ideg + N);
  float*    ndst = nsrc + N;

  hipMemsetAsync(d_ws, 0, (size_t)2 * N * sizeof(unsigned), stream);
  hipMemsetAsync(d_out, 0, (size_t)out_size * sizeof(float), stream);

  deg_kernel <<<(E + 255) / 256, 256, 0, stream>>>(src, dst, odeg, ideg, E);
  norm_kernel<<<(N + 255) / 256, 256, 0, stream>>>(odeg, ideg, nsrc, ndst, N);

  // 4096 blocks * 8 waves = 32768 waves -> ~30 edges/wave for E=1M
  scatter_kernel<<<4096, 256, 0, stream>>>(feat, src, dst, nsrc, out, E);

  const long total = (long)N * 32;                 // one float2 per thread
  scale_kernel<<<(int)((total + 255) / 256), 256, 0, stream>>>(out, ndst, N);
}